// PointerGuidance_88527865905479
// MI455X (gfx1250) — compile-verified
//
#include <hip/hip_runtime.h>
#include <hip/hip_bf16.h>
#include <math.h>

typedef __attribute__((ext_vector_type(2))) float v2f;
typedef __attribute__((ext_vector_type(8))) float v8f;

#define Bsz 32
#define Csz 256
#define Hsz 64
#define Wsz 64
#define HW  (Hsz * Wsz)          // 4096
#define HW4 (HW / 4)             // 1024

__device__ __forceinline__ float clamp01(float x) {
    return fminf(fmaxf(x, 0.0f), 1.0f);
}
__device__ __forceinline__ float sigmoidf_(float x) {
    return 1.0f / (1.0f + __expf(-x));
}

// ---------------------------------------------------------------------------
// Kernel 1: per-batch box math -> mask -> fused conv1(3x3,1->16,relu) +
// conv2(1x1,16->1,sigmoid) -> refined_mask + mask_area. One block per batch.
// Also copies lang_vec into fusion[:,256:512].
// ---------------------------------------------------------------------------
__global__ void k_mask_conv(const float* __restrict__ xywh,
                            const float* __restrict__ lang,
                            const float* __restrict__ c1w,
                            const float* __restrict__ c1b,
                            const float* __restrict__ c2w,
                            const float* __restrict__ c2b,
                            float* __restrict__ boxes_out,
                            float* __restrict__ rmask_out,
                            float* __restrict__ area_ws,
                            float* __restrict__ fusion_ws) {
    const int b = blockIdx.x;
    const int tid = threadIdx.x;

    __shared__ float sm[HW];       // binary mask
    __shared__ float w1[16 * 9];
    __shared__ float bias1[16];
    __shared__ float w2[16];
    __shared__ float red[256];
    __shared__ float sbox[4];      // mx1, my1, mx2, my2

    if (tid == 0) {
        float xc = xywh[b * 4 + 0], yc = xywh[b * 4 + 1];
        float w  = xywh[b * 4 + 2], h  = xywh[b * 4 + 3];
        float x1 = clamp01(xc - 0.5f * w), y1 = clamp01(yc - 0.5f * h);
        float x2 = clamp01(xc + 0.5f * w), y2 = clamp01(yc + 0.5f * h);
        float xl = fminf(x1, x2), xh = fmaxf(x1, x2);
        float yl = fminf(y1, y2), yh = fmaxf(y1, y2);
        float ww = fmaxf(xh - xl, 1e-6f), hh = fmaxf(yh - yl, 1e-6f);
        float cx = 0.5f * (xh + xl),  cy = 0.5f * (yh + yl);
        float bx1 = clamp01(cx - 0.5f * ww), by1 = clamp01(cy - 0.5f * hh);
        float bx2 = clamp01(cx + 0.5f * ww), by2 = clamp01(cy + 0.5f * hh);
        boxes_out[b * 4 + 0] = bx1; boxes_out[b * 4 + 1] = by1;
        boxes_out[b * 4 + 2] = bx2; boxes_out[b * 4 + 3] = by2;
        float bw = fmaxf(bx2 - bx1, 1e-4f), bh = fmaxf(by2 - by1, 1e-4f);
        float diag = sqrtf(bw * bw + bh * bh);
        float margin = fminf(fmaxf(diag * 0.2f, 0.02f), 0.2f);
        sbox[0] = clamp01(bx1 - margin); sbox[1] = clamp01(by1 - margin);
        sbox[2] = clamp01(bx2 + margin); sbox[3] = clamp01(by2 + margin);
    }
    if (tid < 144) w1[tid] = c1w[tid];
    if (tid < 16) { bias1[tid] = c1b[tid]; w2[tid] = c2w[tid]; }
    // fusion[:,256:512] = lang_vec
    fusion_ws[b * 512 + 256 + tid] = lang[b * 256 + tid];
    __syncthreads();

    const float mx1 = sbox[0], my1 = sbox[1], mx2 = sbox[2], my2 = sbox[3];
    for (int i = tid; i < HW; i += 256) {
        int y = i >> 6, x = i & 63;
        float xf = x * (1.0f / 63.0f);
        float yf = y * (1.0f / 63.0f);
        sm[i] = (xf >= mx1 && xf <= mx2 && yf >= my1 && yf <= my2) ? 1.0f : 0.0f;
    }
    __syncthreads();

    const float c2bias = c2b[0];
    float local = 0.0f;
    for (int i = tid; i < HW; i += 256) {
        int y = i >> 6, x = i & 63;
        float acc2 = c2bias;
        #pragma unroll
        for (int c = 0; c < 16; ++c) {
            float s = bias1[c];
            #pragma unroll
            for (int ky = 0; ky < 3; ++ky) {
                int yy = y + ky - 1;
                if (yy < 0 || yy > 63) continue;
                #pragma unroll
                for (int kx = 0; kx < 3; ++kx) {
                    int xx = x + kx - 1;
                    if (xx < 0 || xx > 63) continue;
                    s += w1[c * 9 + ky * 3 + kx] * sm[yy * 64 + xx];
                }
            }
            acc2 += w2[c] * fmaxf(s, 0.0f);
        }
        float r = sigmoidf_(acc2);
        rmask_out[b * HW + i] = r;
        local += r;
    }
    red[tid] = local;
    __syncthreads();
    #pragma unroll
    for (int s = 128; s > 0; s >>= 1) {
        if (tid < s) red[tid] += red[tid + s];
        __syncthreads();
    }
    if (tid == 0) area_ws[b] = fmaxf(red[0], 1.0f);
}

// ---------------------------------------------------------------------------
// Kernel 2: pointer_feat[b,c] = sum_p feat[b,c,p] * rmask[b,p] / area[b]
// One block (256 threads) per (b,c) pair. float4 (b128) streaming loads —
// this pass moves 128 MB of feat and is the HBM roofline term.
// ---------------------------------------------------------------------------
__global__ void k_pointer_feat(const float* __restrict__ feat,
                               const float* __restrict__ rmask,
                               const float* __restrict__ area,
                               float* __restrict__ fusion_ws,
                               float* __restrict__ pf_out) {
    const int bc = blockIdx.x;             // b*256 + c
    const int b = bc >> 8, c = bc & 255;
    const int tid = threadIdx.x;
    const float4* fp4 = (const float4*)(feat + (size_t)bc * HW);
    const float4* mp4 = (const float4*)(rmask + (size_t)b * HW);
    float local = 0.0f;
    #pragma unroll
    for (int i = tid; i < HW4; i += 256) {
        float4 f = fp4[i];
        float4 m = mp4[i];
        local += f.x * m.x + f.y * m.y + f.z * m.z + f.w * m.w;
    }
    __shared__ float red[256];
    red[tid] = local;
    __syncthreads();
    #pragma unroll
    for (int s = 128; s > 0; s >>= 1) {
        if (tid < s) red[tid] += red[tid + s];
        __syncthreads();
    }
    if (tid == 0) {
        float v = red[0] / area[b];
        fusion_ws[b * 512 + c] = v;
        pf_out[b * 256 + c] = v;
    }
}

// ---------------------------------------------------------------------------
// Kernel 3: out(32x256) = act( A(32xK) @ W^T(Kx256) + bias ) via
// V_WMMA_F32_16X16X4_F32. One wave per 16x16 output tile.
//   mode 0: relu   (hidden layers)
//   mode 1: sigmoid (ch_gate)
//   mode 2: residual + 0.4*tanh  (guided_lang)
// A-frag (16x4 f32): lanes 0-15 K={0,1}, lanes 16-31 K={2,3} (VGPR 0/1).
// B-frag (4x16 f32): N striped over lanes, same K split.
// ---------------------------------------------------------------------------
__global__ void k_gemm_wmma(const float* __restrict__ A,
                            const float* __restrict__ W,
                            const float* __restrict__ bias,
                            float* __restrict__ out,
                            const float* __restrict__ residual,
                            int Kdim, int mode) {
    const int lane = threadIdx.x;        // 0..31 (wave32)
    const int mt = blockIdx.x & 1;       // M = 32 -> 2 tiles
    const int nt = blockIdx.x >> 1;      // N = 256 -> 16 tiles
    const int m_base = mt * 16, n_base = nt * 16;
    const int half = lane >> 4;          // K sub-pair select
    const int lm = lane & 15;

    v8f acc = {};
    const int abase = (m_base + lm) * Kdim + 2 * half;
    const int bbase = (n_base + lm) * Kdim + 2 * half;   // W is (256,K): B[k][n]=W[n][k]

    for (int k = 0; k < Kdim; k += 4) {
        v2f a, bf;
        a.x  = A[abase + k];
        a.y  = A[abase + k + 1];
        bf.x = W[bbase + k];
        bf.y = W[bbase + k + 1];
        acc = __builtin_amdgcn_wmma_f32_16x16x4_f32(
            /*neg_a=*/false, a, /*neg_b=*/false, bf,
            /*c_mod=*/(short)0, acc, /*reuse_a=*/false, /*reuse_b=*/false);
    }

    const int col = n_base + lm;
    const float bv = bias[col];
    #pragma unroll
    for (int v = 0; v < 8; ++v) {
        int row = m_base + half * 8 + v;
        float x = acc[v] + bv;
        float y;
        if (mode == 0)      y = fmaxf(x, 0.0f);
        else if (mode == 1) y = sigmoidf_(x);
        else                y = residual[row * 256 + col] + 0.4f * tanhf(x);
        out[row * 256 + col] = y;
    }
}

// ---------------------------------------------------------------------------
// Kernel 4: guided_feat = feat * (1 + 0.6*refined_mask) * (1 + 0.5*ch_gate)
// float4-vectorized; 8,388,608 float4s total.
// ---------------------------------------------------------------------------
__global__ void k_guided_feat(const float* __restrict__ feat,
                              const float* __restrict__ rmask,
                              const float* __restrict__ gate,
                              float* __restrict__ out) {
    size_t idx = (size_t)blockIdx.x * blockDim.x + threadIdx.x; // float4 index
    const size_t total = (size_t)Bsz * Csz * HW4;
    if (idx >= total) return;
    int p4 = (int)(idx & (HW4 - 1));
    int bc = (int)(idx >> 10);           // b*256 + c
    int b = bc >> 8;
    float g = 1.0f + 0.5f * gate[bc];
    float4 f = ((const float4*)feat)[idx];
    float4 m = ((const float4*)rmask)[(size_t)b * HW4 + p4];
    float4 o;
    o.x = f.x * (1.0f + 0.6f * m.x) * g;
    o.y = f.y * (1.0f + 0.6f * m.y) * g;
    o.z = f.z * (1.0f + 0.6f * m.z) * g;
    o.w = f.w * (1.0f + 0.6f * m.w) * g;
    ((float4*)out)[idx] = o;
}

// ---------------------------------------------------------------------------
extern "C" void kernel_launch(void* const* d_in, const int* in_sizes, int n_in,
                              void* d_out, int out_size, void* d_ws, size_t ws_size,
                              hipStream_t stream) {
    const float* feat     = (const float*)d_in[0];
    const float* lang_vec = (const float*)d_in[1];
    const float* xywh     = (const float*)d_in[2];
    const float* conv1_w  = (const float*)d_in[3];
    const float* conv1_b  = (const float*)d_in[4];
    const float* conv2_w  = (const float*)d_in[5];
    const float* conv2_b  = (const float*)d_in[6];
    const float* cg_w1    = (const float*)d_in[7];
    const float* cg_b1    = (const float*)d_in[8];
    const float* cg_w2    = (const float*)d_in[9];
    const float* cg_b2    = (const float*)d_in[10];
    const float* lf_w1    = (const float*)d_in[11];
    const float* lf_b1    = (const float*)d_in[12];
    const float* lf_w2    = (const float*)d_in[13];
    const float* lf_b2    = (const float*)d_in[14];

    float* out = (float*)d_out;
    // output layout (flat, return order)
    float* guided_feat = out;                                   // 32*256*64*64
    float* guided_lang = guided_feat + (size_t)Bsz * Csz * HW;  // 32*256
    float* boxes_xyxy  = guided_lang + Bsz * Csz;               // 32*4
    float* refined_msk = boxes_xyxy + Bsz * 4;                  // 32*64*64
    float* pointer_ft  = refined_msk + (size_t)Bsz * HW;        // 32*256

    // workspace layout
    float* ws      = (float*)d_ws;
    float* fusion  = ws;                 // 32*512 = 16384
    float* area    = fusion + 32 * 512;  // 32
    float* hidden  = area + 32;          // 32*256 = 8192 (reused for both MLPs)
    float* ch_gate = hidden + 32 * 256;  // 32*256 = 8192

    // 1. box -> mask -> conv -> refined_mask + area; lang half of fusion
    k_mask_conv<<<Bsz, 256, 0, stream>>>(xywh, lang_vec, conv1_w, conv1_b,
                                         conv2_w, conv2_b, boxes_xyxy,
                                         refined_msk, area, fusion);

    // 2. pointer_feat (masked mean of feat, b128 streaming)
    k_pointer_feat<<<Bsz * Csz, 256, 0, stream>>>(feat, refined_msk, area,
                                                  fusion, pointer_ft);

    // 3. MLPs via f32 WMMA (one wave per 16x16 tile; 2x16 tiles)
    k_gemm_wmma<<<32, 32, 0, stream>>>(fusion, cg_w1, cg_b1, hidden, nullptr, 512, 0);
    k_gemm_wmma<<<32, 32, 0, stream>>>(hidden, cg_w2, cg_b2, ch_gate, nullptr, 256, 1);
    k_gemm_wmma<<<32, 32, 0, stream>>>(fusion, lf_w1, lf_b1, hidden, nullptr, 512, 0);
    k_gemm_wmma<<<32, 32, 0, stream>>>(hidden, lf_w2, lf_b2, guided_lang, lang_vec, 256, 2);

    // 4. guided_feat elementwise (float4 vectorized)
    const int total4 = Bsz * Csz * HW4;             // 8,388,608
    k_guided_feat<<<total4 / 256, 256, 0, stream>>>(feat, refined_msk, ch_gate,
                                                    guided_feat);
}